// instance_loss_52338471469341
// MI455X (gfx1250) — compile-verified
//
#include <hip/hip_runtime.h>
#include <hip/hip_bf16.h>
#include <math.h>

typedef __attribute__((ext_vector_type(2))) float v2f;
typedef __attribute__((ext_vector_type(8))) float v8f;

#define BATCH  8
#define CDIM   128
#define NDIM   64
#define PDIM   65536
#define KSLABS 64
#define KSLAB  (PDIM / KSLABS)   // 1024

// ---------------------------------------------------------------------------
// Kernel 0: zero accumulators + output (workspace/d_out are poisoned by harness)
// ---------------------------------------------------------------------------
__global__ __launch_bounds__(256) void il_zero_kernel(float* __restrict__ cs,
                                                      float* __restrict__ as,
                                                      float* __restrict__ out) {
  int i = blockIdx.x * blockDim.x + threadIdx.x;
  if (i < BATCH * CDIM * NDIM) cs[i] = 0.0f;
  if (i < BATCH * CDIM)        as[i] = 0.0f;
  if (i == 0)                  out[0] = 0.0f;
}

// ---------------------------------------------------------------------------
// Kernel 1: fused cross-GEMM (f32 WMMA 16x16x4) + softplus row reduction.
// One wave per (batch, 16-row tile of C, K-slab). A fragment reused across the
// four 16-column tiles of N. Layouts per CDNA5 ISA 7.12.2:
//   A 16x4 f32 : lanes 0-15 -> M=lane, VGPR0/1 = K {0,1}; lanes 16-31 -> K {2,3}
//   B 4x16 f32 : lanes 0-15 -> N=lane, VGPR0/1 = K {0,1}; lanes 16-31 -> K {2,3}
//   D 16x16 f32: VGPR r, lanes 0-15 -> M=r, N=lane; lanes 16-31 -> M=r+8
// Softplus uses __expf/__logf (single TRANS ops that co-execute with WMMA);
// |err| vs log1p ~1e-7 absolute, negligible in a 65536-term mean.
// X is streamed with non-temporal loads (read exactly once grid-wide) so L2
// stays dedicated to the T rows that are re-read by all row-tiles/K-slabs.
// ---------------------------------------------------------------------------
__global__ __launch_bounds__(32) void il_cost_gemm_wmma(
    const float* __restrict__ X,      // [B, C, P] logits
    const float* __restrict__ T,      // [B, N, P] binary targets
    float* __restrict__ crossSum,     // [B, C, N] (atomic accumulated)
    float* __restrict__ aSum)         // [B, C]    (atomic accumulated)
{
  const int slab = blockIdx.x;            // 0..KSLABS-1
  const int rt   = blockIdx.y;            // 0..7  : C row tile
  const int b    = blockIdx.z;            // 0..7  : batch
  const int lane = threadIdx.x;
  const int ln   = lane & 15;
  const int koff = (lane >> 4) * 2;       // half-wave K split

  const float* Xb = X + ((size_t)b * CDIM + (size_t)rt * 16 + ln) * PDIM;
  const float* Tb = T + ((size_t)b * NDIM + ln) * PDIM;

  v8f acc0 = {}, acc1 = {}, acc2 = {}, acc3 = {};
  float sp = 0.0f;

  const int k0 = slab * KSLAB;
  const int k1 = k0 + KSLAB;
#pragma unroll 4
  for (int k = k0; k < k1; k += 4) {
    // A fragment: one NT load per wave, reused for all 4 column tiles.
    v2f a = __builtin_nontemporal_load((const v2f*)(Xb + k + koff));

    // Fused softplus: max(x,0) + log(1 + exp(-|x|)) with fast TRANS ops.
    sp += fmaxf(a.x, 0.0f) + __logf(1.0f + __expf(-fabsf(a.x)));
    sp += fmaxf(a.y, 0.0f) + __logf(1.0f + __expf(-fabsf(a.y)));

    v2f b0 = *(const v2f*)(Tb + (size_t) 0 * PDIM + k + koff);
    v2f b1 = *(const v2f*)(Tb + (size_t)16 * PDIM + k + koff);
    v2f b2 = *(const v2f*)(Tb + (size_t)32 * PDIM + k + koff);
    v2f b3 = *(const v2f*)(Tb + (size_t)48 * PDIM + k + koff);

    acc0 = __builtin_amdgcn_wmma_f32_16x16x4_f32(false, a, false, b0, (short)0, acc0, false, false);
    acc1 = __builtin_amdgcn_wmma_f32_16x16x4_f32(false, a, false, b1, (short)0, acc1, false, false);
    acc2 = __builtin_amdgcn_wmma_f32_16x16x4_f32(false, a, false, b2, (short)0, acc2, false, false);
    acc3 = __builtin_amdgcn_wmma_f32_16x16x4_f32(false, a, false, b3, (short)0, acc3, false, false);
  }

  // Scatter partial GEMM results.
  const int mbase = rt * 16 + (lane >> 4) * 8;
  float* cs = crossSum + (size_t)b * CDIM * NDIM;
#pragma unroll
  for (int r = 0; r < 8; ++r) {
    const size_t row = (size_t)(mbase + r) * NDIM;
    __hip_atomic_fetch_add(&cs[row +  0 + ln], acc0[r], __ATOMIC_RELAXED, __HIP_MEMORY_SCOPE_AGENT);
    __hip_atomic_fetch_add(&cs[row + 16 + ln], acc1[r], __ATOMIC_RELAXED, __HIP_MEMORY_SCOPE_AGENT);
    __hip_atomic_fetch_add(&cs[row + 32 + ln], acc2[r], __ATOMIC_RELAXED, __HIP_MEMORY_SCOPE_AGENT);
    __hip_atomic_fetch_add(&cs[row + 48 + ln], acc3[r], __ATOMIC_RELAXED, __HIP_MEMORY_SCOPE_AGENT);
  }

  // Softplus partials: lane l and l+16 hold the two K-halves of row ln.
  float spo = __shfl_xor(sp, 16, 32);
  if (lane < 16) {
    __hip_atomic_fetch_add(&aSum[b * CDIM + rt * 16 + ln], sp + spo,
                           __ATOMIC_RELAXED, __HIP_MEMORY_SCOPE_AGENT);
  }
}

// ---------------------------------------------------------------------------
// Kernel 2: cost[b,c,n] = aSum[b,c]/P - crossSum[b,c,n]/P   (in place)
// ---------------------------------------------------------------------------
__global__ __launch_bounds__(256) void il_finalize_kernel(float* __restrict__ cross,
                                                          const float* __restrict__ aSum) {
  int idx = blockIdx.x * blockDim.x + threadIdx.x;
  if (idx >= BATCH * CDIM * NDIM) return;
  int c = (idx / NDIM) % CDIM;
  int b = idx / (NDIM * CDIM);
  const float invP = 1.0f / (float)PDIM;
  cross[idx] = aSum[b * CDIM + c] * invP - cross[idx] * invP;
}

// ---------------------------------------------------------------------------
// Kernel 3: wave-parallel Jonker-Volgenant assignment, one wave per batch.
// Works on the transposed matrix (n=64 rows, m=128 cols) exactly like the
// reference (cost.shape[0] > cost.shape[1] -> transpose). Column scans and
// the free-column argmin are lane-parallel (4 columns / lane), with a
// lexicographic (value, index) shfl reduction to replicate np.argmin ties.
// ---------------------------------------------------------------------------
#define HN 64    // rows of transposed problem  (original N)
#define HM 128   // cols of transposed problem  (original C)

__global__ __launch_bounds__(32) void il_hungarian_kernel(const float* __restrict__ cost,
                                                          float* __restrict__ out) {
  const int b    = blockIdx.x;
  const int lane = threadIdx.x;

  __shared__ float ct[HN][HM];      // ct[i][j] = cost[b, j, i]
  __shared__ float v[HM + 1];
  __shared__ float minv[HM + 1];
  __shared__ float u[HN + 1];
  __shared__ int   p[HM + 1];
  __shared__ int   way[HM + 1];
  __shared__ int   used[HM + 1];

  const float* cb = cost + (size_t)b * CDIM * NDIM;   // [C=128][N=64]
  for (int idx = lane; idx < HN * HM; idx += 32) {
    int i = idx / HM, j = idx % HM;
    ct[i][j] = cb[(size_t)j * NDIM + i];
  }
  for (int j = lane; j <= HM; j += 32) { v[j] = 0.0f; p[j] = 0; way[j] = 0; }
  for (int i = lane; i <= HN; i += 32) u[i] = 0.0f;
  __syncthreads();

  const float INF = 1e30f;
  for (int i = 1; i <= HN; ++i) {
    if (lane == 0) p[0] = i;
    for (int j = lane; j <= HM; j += 32) { minv[j] = INF; used[j] = 0; }
    __syncthreads();

    int j0 = 0;
    int jfin;
    for (;;) {
      if (lane == 0) used[j0] = 1;
      __syncthreads();
      const int   i0  = p[j0];
      const float ui0 = u[i0];

      // Lane-parallel: relax minv and find argmin over free columns.
      float best  = INF;
      int   bestj = -1;
      for (int jj = lane + 1; jj <= HM; jj += 32) {
        if (!used[jj]) {
          float cur = ct[i0 - 1][jj - 1] - ui0 - v[jj];
          if (cur < minv[jj]) { minv[jj] = cur; way[jj] = j0; }
          if (minv[jj] < best) { best = minv[jj]; bestj = jj; }
        }
      }
      // (value, index)-lexicographic min across the wave (np.argmin ties).
      for (int off = 16; off > 0; off >>= 1) {
        float ob = __shfl_xor(best, off, 32);
        int   oj = __shfl_xor(bestj, off, 32);
        if (ob < best || (ob == best && (unsigned)oj < (unsigned)bestj)) {
          best = ob; bestj = oj;
        }
      }
      const float delta = best;
      const int   j1    = bestj;
      __syncthreads();   // scans done before mutating u/v/minv

      for (int jj = lane; jj <= HM; jj += 32) {
        if (used[jj]) { u[p[jj]] += delta; v[jj] -= delta; }
        else          { minv[jj] -= delta; }
      }
      __syncthreads();

      j0 = j1;
      if (p[j0] == 0) { jfin = j0; break; }
    }

    if (lane == 0) {            // augmenting path flip (cheap, serial)
      int j0a = jfin;
      while (j0a != 0) { int j1a = way[j0a]; p[j0a] = p[j1a]; j0a = j1a; }
    }
    __syncthreads();
  }

  // Sum assigned costs: pair (C-row j-1, N-col p[j]-1) where p[j] != 0.
  float t = 0.0f;
  for (int j = lane + 1; j <= HM; j += 32)
    if (p[j] != 0) t += ct[p[j] - 1][j - 1];
  for (int off = 16; off > 0; off >>= 1) t += __shfl_xor(t, off, 32);
  if (lane == 0) {
    __hip_atomic_fetch_add(out, t / (float)(HN * BATCH),
                           __ATOMIC_RELAXED, __HIP_MEMORY_SCOPE_AGENT);
  }
}

// ---------------------------------------------------------------------------
extern "C" void kernel_launch(void* const* d_in, const int* in_sizes, int n_in,
                              void* d_out, int out_size, void* d_ws, size_t ws_size,
                              hipStream_t stream) {
  const float* X = (const float*)d_in[0];   // pred_masks [8,128,256,256] f32
  const float* T = (const float*)d_in[1];   // gt_masks   [8,64,256,256]  f32
  float* out = (float*)d_out;               // scalar f32

  float* crossSum = (float*)d_ws;                         // 65536 floats
  float* aSum     = crossSum + BATCH * CDIM * NDIM;       // 1024 floats

  const int NC = BATCH * CDIM * NDIM;
  il_zero_kernel<<<(NC + 255) / 256, 256, 0, stream>>>(crossSum, aSum, out);

  dim3 g(KSLABS, CDIM / 16, BATCH);   // 64 x 8 x 8 waves
  il_cost_gemm_wmma<<<g, 32, 0, stream>>>(X, T, crossSum, aSum);

  il_finalize_kernel<<<(NC + 255) / 256, 256, 0, stream>>>(crossSum, aSum);

  il_hungarian_kernel<<<BATCH, 32, 0, stream>>>(crossSum, out);
}